// MMD_25065429139928
// MI455X (gfx1250) — compile-verified
//
#include <hip/hip_runtime.h>
#include <hip/hip_bf16.h>

typedef __attribute__((ext_vector_type(2))) float v2f;
typedef __attribute__((ext_vector_type(4))) float v4f;
typedef __attribute__((ext_vector_type(8))) float v8f;

#define N_PTS 2048
#define D_DIM 512
#define N_SIG 19

// betas = 1 / (2 * sigma)
__constant__ float c_betas[N_SIG] = {
    0.5f / 1e-06f, 0.5f / 1e-05f, 0.5f / 1e-04f, 0.5f / 1e-03f, 0.5f / 1e-02f,
    0.5f / 1e-01f, 0.5f / 1.0f,   0.5f / 5.0f,   0.5f / 10.0f,  0.5f / 15.0f,
    0.5f / 20.0f,  0.5f / 25.0f,  0.5f / 30.0f,  0.5f / 35.0f,  0.5f / 100.0f,
    0.5f / 1e3f,   0.5f / 1e4f,   0.5f / 1e5f,   0.5f / 1e6f};

// ---------------------------------------------------------------------------
// Kernel 1: row squared-norms for hs and ht.
// grid.x in [0, 4096): low 2048 -> hs rows, high 2048 -> ht rows.
// ---------------------------------------------------------------------------
__global__ void mmd_row_norms(const float* __restrict__ hs,
                              const float* __restrict__ ht,
                              float* __restrict__ norms) {
  __shared__ float red[256];
  const int tid = threadIdx.x;
  const int which = blockIdx.x >> 11;         // 0 = hs, 1 = ht
  const int row = blockIdx.x & (N_PTS - 1);
  const float* __restrict__ X = which ? ht : hs;
  const float* __restrict__ r = X + (size_t)row * D_DIM;

  float a = r[tid];
  float b = r[tid + 256];
  red[tid] = a * a + b * b;
  __syncthreads();
#pragma unroll
  for (int s = 128; s > 0; s >>= 1) {
    if (tid < s) red[tid] += red[tid + s];
    __syncthreads();
  }
  if (tid == 0) norms[which * N_PTS + row] = red[0];
}

// ---------------------------------------------------------------------------
// Kernel 2: fused Gram-tile GEMM (fp32 WMMA 16x16x4) + multi-sigma Gaussian
// kernel-sum epilogue. Workgroup = 256 threads (8 waves), tile = 128x64.
// blockIdx.z selects the matrix pair: 0:(hs,hs,+1) 1:(ht,ht,+1) 2:(hs,ht,-2).
// Writes one weighted partial sum per workgroup into `partials`.
// ---------------------------------------------------------------------------
#define TILE_M 128
#define TILE_N 64
#define KC 64
#define LDSP 68  // LDS row stride in floats: 16B aligned, bank-staggered

__global__ void mmd_gram_fused(const float* __restrict__ hs,
                               const float* __restrict__ ht,
                               const float* __restrict__ norms,
                               float* __restrict__ partials) {
  __shared__ float Xs[TILE_M][LDSP];
  __shared__ float Ys[TILE_N][LDSP];
  __shared__ float nX[TILE_M];
  __shared__ float nY[TILE_N];
  __shared__ float red[256];

  const int tid = threadIdx.x;
  const int wave = tid >> 5;
  const int lane = tid & 31;
  const int l15 = lane & 15;
  const int hi = lane >> 4;

  const int tileN = blockIdx.x * TILE_N;
  const int tileM = blockIdx.y * TILE_M;
  const int z = blockIdx.z;

  const float* __restrict__ Xp;
  const float* __restrict__ Yp;
  const float* __restrict__ nxp;
  const float* __restrict__ nyp;
  float weight;
  if (z == 0) {
    Xp = hs; Yp = hs; nxp = norms; nyp = norms; weight = 1.0f;
  } else if (z == 1) {
    Xp = ht; Yp = ht; nxp = norms + N_PTS; nyp = norms + N_PTS; weight = 1.0f;
  } else {
    Xp = hs; Yp = ht; nxp = norms; nyp = norms + N_PTS; weight = -2.0f;
  }

  // Stage the norm slices for this tile.
  if (tid < TILE_M) {
    nX[tid] = nxp[tileM + tid];
  } else if (tid < TILE_M + TILE_N) {
    nY[tid - TILE_M] = nyp[tileN + (tid - TILE_M)];
  }

  v8f acc[4];
#pragma unroll
  for (int c = 0; c < 4; ++c) acc[c] = (v8f){0.f, 0.f, 0.f, 0.f, 0.f, 0.f, 0.f, 0.f};

  for (int k0 = 0; k0 < D_DIM; k0 += KC) {
    __syncthreads();  // protect LDS reuse (also covers the norm staging)
    // Load X chunk: 128 rows x 64 cols = 2048 float4, 8 per thread.
#pragma unroll
    for (int i = 0; i < 8; ++i) {
      int idx = tid + i * 256;
      int row = idx >> 4;
      int c4 = (idx & 15) << 2;
      v4f v = *(const v4f*)(Xp + (size_t)(tileM + row) * D_DIM + k0 + c4);
      *(v4f*)&Xs[row][c4] = v;
    }
    // Load Y chunk: 64 rows x 64 cols = 1024 float4, 4 per thread.
#pragma unroll
    for (int i = 0; i < 4; ++i) {
      int idx = tid + i * 256;
      int row = idx >> 4;
      int c4 = (idx & 15) << 2;
      v4f v = *(const v4f*)(Yp + (size_t)(tileN + row) * D_DIM + k0 + c4);
      *(v4f*)&Ys[row][c4] = v;
    }
    __syncthreads();

    // 16 k-steps of 4; each wave owns rows [wave*16, wave*16+16).
#pragma unroll
    for (int kk = 0; kk < KC; kk += 4) {
      // A fragment (16x4 fp32): lanes 0-15 hold K=k,k+1; lanes 16-31 K=k+2,k+3.
      v2f a = *(const v2f*)&Xs[wave * 16 + l15][kk + 2 * hi];
#pragma unroll
      for (int c = 0; c < 4; ++c) {
        // B fragment (4x16 fp32): B[k][n] = Y[n][k], same lane split as A.
        v2f b = *(const v2f*)&Ys[c * 16 + l15][kk + 2 * hi];
        acc[c] = __builtin_amdgcn_wmma_f32_16x16x4_f32(
            false, a, false, b, (short)0, acc[c], false, false);
      }
    }
  }

  // Epilogue: dist = ||x||^2 + ||y||^2 - 2*dot, clamp >= 0 (exact for a
  // squared distance; keeps exp(-5e5 * dist) sane on the diagonal), then
  // sum exp(-beta_s * dist) over the 19 sigmas.
  float lsum = 0.f;
#pragma unroll
  for (int c = 0; c < 4; ++c) {
#pragma unroll
    for (int e = 0; e < 8; ++e) {
      int mi = wave * 16 + (hi ? e + 8 : e);  // C/D layout: VGPR e -> M
      int nj = c * 16 + l15;                  // lane -> N
      float dot = acc[c][e];
      float dist = nX[mi] + nY[nj] - 2.0f * dot;
      dist = fmaxf(dist, 0.0f);
      float s = 0.f;
#pragma unroll
      for (int t = 0; t < N_SIG; ++t) s += __expf(-c_betas[t] * dist);
      lsum += s;
    }
  }

  red[tid] = lsum;
  __syncthreads();
#pragma unroll
  for (int s = 128; s > 0; s >>= 1) {
    if (tid < s) red[tid] += red[tid + s];
    __syncthreads();
  }
  if (tid == 0) {
    int pidx = z * (gridDim.x * gridDim.y) + blockIdx.y * gridDim.x + blockIdx.x;
    partials[pidx] = weight * red[0];
  }
}

// ---------------------------------------------------------------------------
// Kernel 3: deterministic final reduction + clamps.
// ---------------------------------------------------------------------------
__global__ void mmd_finalize(const float* __restrict__ partials, int n_part,
                             float* __restrict__ out) {
  __shared__ float red[256];
  const int tid = threadIdx.x;
  float s = 0.f;
  for (int i = tid; i < n_part; i += 256) s += partials[i];
  red[tid] = s;
  __syncthreads();
#pragma unroll
  for (int k = 128; k > 0; k >>= 1) {
    if (tid < k) red[tid] += red[tid + k];
    __syncthreads();
  }
  if (tid == 0) {
    float cost = red[0] / ((float)N_PTS * (float)N_PTS);
    cost = fmaxf(cost, 0.0f);
    cost = fmaxf(cost, 1e-4f);
    out[0] = cost;
  }
}

// ---------------------------------------------------------------------------
extern "C" void kernel_launch(void* const* d_in, const int* in_sizes, int n_in,
                              void* d_out, int out_size, void* d_ws, size_t ws_size,
                              hipStream_t stream) {
  (void)in_sizes; (void)n_in; (void)out_size; (void)ws_size;
  const float* hs = (const float*)d_in[0];
  const float* ht = (const float*)d_in[1];
  float* out = (float*)d_out;

  // Workspace layout (floats): [0,2048) norms(hs), [2048,4096) norms(ht),
  // [4096, 4096+1536) per-workgroup partial sums.
  float* norms = (float*)d_ws;
  float* partials = norms + 2 * N_PTS;
  const int n_part = 32 * 16 * 3;  // gridDim.x * gridDim.y * gridDim.z

  mmd_row_norms<<<dim3(2 * N_PTS), dim3(256), 0, stream>>>(hs, ht, norms);

  dim3 grid(N_PTS / TILE_N, N_PTS / TILE_M, 3);  // (32, 16, 3)
  mmd_gram_fused<<<grid, dim3(256), 0, stream>>>(hs, ht, norms, partials);

  mmd_finalize<<<dim3(1), dim3(256), 0, stream>>>(partials, n_part, out);
}